// Point_Transformer_12446815224298
// MI455X (gfx1250) — compile-verified
//
#include <hip/hip_runtime.h>
#include <math.h>

// Problem constants (match reference)
#define B_   4
#define N_   4096
#define M_   4096
#define C_   64
#define K_   32
#define RAD2 0.01f      // RADIUS^2
#define INVR 10.0f      // 1/RADIUS
#define EPS_ 1e-5f

#define LSTR 68         // LDS column stride (floats): 68 mod 64 = 4 -> conflict-free b64 reads

typedef __attribute__((ext_vector_type(2))) float v2f;
typedef __attribute__((ext_vector_type(8))) float v8f;
typedef __attribute__((ext_vector_type(4))) int   v4i;

// CDNA5 async global->LDS gather (ASYNCcnt-tracked), with safe fallback.
#if defined(__gfx1250__) && __has_builtin(__builtin_amdgcn_global_load_async_to_lds_b128) && \
    __has_builtin(__builtin_amdgcn_s_wait_asynccnt)
#define USE_ASYNC_GATHER 1
typedef __attribute__((address_space(1))) v4i* gas1_v4i;
typedef __attribute__((address_space(3))) v4i* las3_v4i;
#endif

// ---------------------------------------------------------------------------
// Kernel 0: transpose support_features [B,C,M] -> featT [B,M,C]
// ---------------------------------------------------------------------------
__global__ void __launch_bounds__(256) transpose_feat(const float* __restrict__ src,
                                                      float* __restrict__ dst) {
  __shared__ float tile[32][33];
  const int b  = blockIdx.z;
  const int m0 = blockIdx.x * 32;
  const int c0 = blockIdx.y * 32;
  const int tx = threadIdx.x, ty = threadIdx.y;  // block (32,8)
#pragma unroll
  for (int i = 0; i < 32; i += 8)
    tile[ty + i][tx] = src[((size_t)b * C_ + (c0 + ty + i)) * M_ + (m0 + tx)];
  __syncthreads();
#pragma unroll
  for (int i = 0; i < 32; i += 8)
    dst[((size_t)b * M_ + (m0 + ty + i)) * C_ + (c0 + tx)] = tile[tx][ty + i];
}

// ---------------------------------------------------------------------------
// Kernel 1: ball query = exact top-K(32) nearest (ascending d2), wave32/query.
// Outputs idx[B*N,K] and selected d2[B*N,K]; idx[...,0] == argmin == x_i index.
// ---------------------------------------------------------------------------
__global__ void __launch_bounds__(128) ball_query_kernel(const float* __restrict__ qxyz,
                                                         const float* __restrict__ sxyz,
                                                         const int*   __restrict__ smask,
                                                         int*   __restrict__ idxbuf,
                                                         float* __restrict__ d2buf) {
  extern __shared__ float smem[];
  float* sp = smem;                          // [M][4] = x,y,z,maskf  (64 KB)
  const int tid  = threadIdx.x;
  const int lane = tid & 31;
  const int wave = tid >> 5;
  const int q    = blockIdx.x * 4 + wave;    // global query id (4 waves/block)
  const int b    = q / N_;

  // stage support points + mask for this batch (all queries in block share b)
  for (int m = tid; m < M_; m += 128) {
    const size_t sb = (size_t)b * M_ + m;
    sp[m * 4 + 0] = sxyz[sb * 3 + 0];
    sp[m * 4 + 1] = sxyz[sb * 3 + 1];
    sp[m * 4 + 2] = sxyz[sb * 3 + 2];
    sp[m * 4 + 3] = (smask[sb] > 0) ? 1.f : 0.f;
  }
  __syncthreads();

  float* d2w = smem + M_ * 4 + wave * M_;    // per-wave d2 row (16 KB each)
  const float qx = qxyz[(size_t)q * 3 + 0];
  const float qy = qxyz[(size_t)q * 3 + 1];
  const float qz = qxyz[(size_t)q * 3 + 2];

  for (int m = lane; m < M_; m += 32) {
    const float dx = qx - sp[m * 4 + 0];
    const float dy = qy - sp[m * 4 + 1];
    const float dz = qz - sp[m * 4 + 2];
    const float d  = dx * dx + dy * dy + dz * dz;
    d2w[m] = (sp[m * 4 + 3] > 0.f) ? d : __builtin_inff();
  }

  int   selIdx = 0;
  float selD2  = __builtin_inff();
#pragma unroll 1
  for (int kr = 0; kr < K_; ++kr) {
    float bv = __builtin_inff();
    int   bi = 0x7fffffff;
    for (int m = lane; m < M_; m += 32) {
      const float d = d2w[m];                 // NaN (= removed) never passes
      if (d < bv || (d == bv && m < bi)) { bv = d; bi = m; }
    }
    // wave32 min-reduction with index tie-break (matches top_k ordering)
#pragma unroll
    for (int off = 16; off >= 1; off >>= 1) {
      const float ov = __shfl_xor(bv, off);
      const int   oi = __shfl_xor(bi, off);
      if (ov < bv || (ov == bv && oi < bi)) { bv = ov; bi = oi; }
    }
    const int mi = (bi < M_) ? bi : 0;
    if (lane == kr) { selIdx = mi; selD2 = bv; }
    // remove selected entry; owner lane (bi&31) writes/reads its own stride -> in-order DS
    if (bi < M_ && lane == (bi & 31)) d2w[bi] = __builtin_nanf("");
  }
  idxbuf[(size_t)q * K_ + lane] = selIdx;
  d2buf [(size_t)q * K_ + lane] = selD2;
}

// ---------------------------------------------------------------------------
// WMMA helper: accumulate D(16x16) += W[o-tile, :64] x X[:64, col-tile]
// using chained V_WMMA_F32_16X16X4_F32 (16 k-steps of 4).
// A lane map (16x4 f32):   m = lane&15, k = 2*(lane>>4) + v
// B lane map (4x16 f32):   n = lane&15, k = 2*(lane>>4) + v
// Both fragments are 8B-aligned pairs -> b64 loads.
// ---------------------------------------------------------------------------
__device__ __forceinline__ v8f wmma_tile64(const float* __restrict__ Wg,
                                           const float* __restrict__ Bl,  // LDS [col][LSTR]
                                           int otile, int ct, int lane) {
  v8f acc = {0.f, 0.f, 0.f, 0.f, 0.f, 0.f, 0.f, 0.f};
  const int m  = lane & 15;
  const int kh = lane >> 4;
  const float* wrow = Wg + (otile * 16 + m) * C_ + 2 * kh;
  const float* bcol = Bl + (ct * 16 + m) * LSTR + 2 * kh;
#pragma unroll
  for (int cb = 0; cb < C_; cb += 4) {
    const v2f A  = *(const v2f*)(wrow + cb);
    const v2f Bv = *(const v2f*)(bcol + cb);
    acc = __builtin_amdgcn_wmma_f32_16x16x4_f32(false, A, false, Bv, (short)0, acc,
                                                false, false);
  }
  return acc;
}

// ---------------------------------------------------------------------------
// Kernel 2: fused point-transformer. One wave per query (32 cols = K nbrs).
// cp layout (64 floats each): 0 b_t1, 1 b_t2, 2 b_psi, 3 b_alpha, 4 b_g1,
//                             5 b_g2, 6 inv_t, 7 sh_t, 8 inv_g, 9 sh_g
// wt1p: zero-padded W_theta1 as [64][4] (removes divergent guarded loads).
// ---------------------------------------------------------------------------
__global__ void __launch_bounds__(128) pt_fused_kernel(
    const float* __restrict__ qxyz, const float* __restrict__ sxyz,
    const int* __restrict__ qmask,
    const float* __restrict__ featT, const int* __restrict__ idxbuf,
    const float* __restrict__ d2buf,
    const float* __restrict__ Wt1, const float* __restrict__ bt1,
    const float* __restrict__ Wt2, const float* __restrict__ bt2,
    const float* __restrict__ Wphi, const float* __restrict__ bphi,
    const float* __restrict__ Wpsi, const float* __restrict__ bpsi,
    const float* __restrict__ Wal,  const float* __restrict__ bal,
    const float* __restrict__ Wg1,  const float* __restrict__ bg1,
    const float* __restrict__ Wg2,  const float* __restrict__ bg2,
    const float* __restrict__ g_t, const float* __restrict__ be_t,
    const float* __restrict__ rm_t, const float* __restrict__ rv_t,
    const float* __restrict__ g_g, const float* __restrict__ be_g,
    const float* __restrict__ rm_g, const float* __restrict__ rv_g,
    float* __restrict__ out) {
  extern __shared__ float smem[];
  const int tid  = threadIdx.x;
  const int lane = tid & 31;
  const int wave = tid >> 5;
  const int q    = blockIdx.x * 4 + wave;
  const int b    = q / N_;
  const int n    = q % N_;
  const int m16  = lane & 15;
  const int kh   = lane >> 4;

  float* cp   = smem;        // 640 floats of channel params, shared by block
  float* wt1p = smem + 640;  // 256 floats: padded W_theta1 [64][4]
  if (tid < C_) {
    const int o = tid;
    cp[0 * C_ + o] = bt1[o];
    cp[1 * C_ + o] = bt2[o];
    cp[2 * C_ + o] = bpsi[o];
    cp[3 * C_ + o] = bal[o];
    cp[4 * C_ + o] = bg1[o];
    cp[5 * C_ + o] = bg2[o];
    const float it = g_t[o] * rsqrtf(rv_t[o] + EPS_);
    cp[6 * C_ + o] = it;
    cp[7 * C_ + o] = be_t[o] - rm_t[o] * it;
    const float ig = g_g[o] * rsqrtf(rv_g[o] + EPS_);
    cp[8 * C_ + o] = ig;
    cp[9 * C_ + o] = be_g[o] - rm_g[o] * ig;
    wt1p[o * 4 + 0] = Wt1[o * 3 + 0];
    wt1p[o * 4 + 1] = Wt1[o * 3 + 1];
    wt1p[o * 4 + 2] = Wt1[o * 3 + 2];
    wt1p[o * 4 + 3] = 0.f;
  }

  float* wb   = smem + 896 + wave * 8928;  // per-wave region
  float* xj   = wb;           // [32][LSTR] gathered neighbor features
  float* tmp  = xj  + 2176;   // [32][LSTR] scratch (theta1 out / gamma1 out / feats)
  float* del  = tmp + 2176;   // [32][LSTR] delta (positional encoding)
  float* rel  = del + 2176;   // [32][LSTR] rel / softmax weights
  float* posb = rel + 2176;   // [32][4] pos (row 3 zero-padded for K=4 WMMA)
  float* lini = posb + 128;   // [64] phi(x_i)+b_phi
  float* fm   = lini + 64;    // [32] per-neighbor feature mask

  // --- per-column setup (col == lane) ------------------------------------
  const float qx = qxyz[(size_t)q * 3 + 0];
  const float qy = qxyz[(size_t)q * 3 + 1];
  const float qz = qxyz[(size_t)q * 3 + 2];
  const int   nidx = idxbuf[(size_t)q * K_ + lane];
  const float d2v  = d2buf [(size_t)q * K_ + lane];
  {
    const size_t sb = (size_t)b * M_ + nidx;
    posb[lane * 4 + 0] = (sxyz[sb * 3 + 0] - qx) * INVR;
    posb[lane * 4 + 1] = (sxyz[sb * 3 + 1] - qy) * INVR;
    posb[lane * 4 + 2] = (sxyz[sb * 3 + 2] - qz) * INVR;
    posb[lane * 4 + 3] = 0.f;
    fm[lane] = ((d2v <= RAD2) ? 1.f : 0.f) + (1.f - (float)qmask[q]);
    // gather x_j: 256B contiguous row per neighbor -> 16 x b128 per lane
    const float* fsrc = featT + sb * C_;
    float* xdst = xj + lane * LSTR;
#if defined(USE_ASYNC_GATHER)
#pragma unroll
    for (int j = 0; j < C_; j += 4) {
      __builtin_amdgcn_global_load_async_to_lds_b128((gas1_v4i)(fsrc + j),
                                                     (las3_v4i)(xdst + j), 0, 0);
    }
#else
#pragma unroll
    for (int j = 0; j < C_; j += 4) {
      float4 t = *(const float4*)(fsrc + j);
      *(float4*)(xdst + j) = t;
    }
#endif
  }

  // --- lin_i = phi(x_i) + b_phi (tiny 64x64 MV, 2 channels/lane) ----------
  // (overlaps with the async x_j gather)
  {
    const int nn = idxbuf[(size_t)q * K_ + 0];  // argmin == round 0
    const float* fi = featT + ((size_t)b * M_ + nn) * C_;
    for (int oo = lane; oo < C_; oo += 32) {
      float s = bphi[oo];
#pragma unroll 4
      for (int c = 0; c < C_; ++c) s += Wphi[oo * C_ + c] * fi[c];
      lini[oo] = s;
    }
  }
#if defined(USE_ASYNC_GATHER)
  __builtin_amdgcn_s_wait_asynccnt(0);  // x_j resident in LDS
#endif
  __syncthreads();

  // --- theta1: tmp = W_theta1(3->64) * pos + b_t1 (one K=4 WMMA per tile) -
  for (int otile = 0; otile < 4; ++otile)
    for (int ct = 0; ct < 2; ++ct) {
      const v2f A  = *(const v2f*)(wt1p + (otile * 16 + m16) * 4 + 2 * kh);
      const v2f Bv = *(const v2f*)(posb + (ct * 16 + m16) * 4 + 2 * kh);
      v8f acc = {0.f, 0.f, 0.f, 0.f, 0.f, 0.f, 0.f, 0.f};
      acc = __builtin_amdgcn_wmma_f32_16x16x4_f32(false, A, false, Bv, (short)0, acc,
                                                  false, false);
#pragma unroll
      for (int r = 0; r < 8; ++r) {
        const int o = otile * 16 + r + 8 * kh;
        tmp[(ct * 16 + m16) * LSTR + o] = acc[r] + cp[0 * C_ + o];
      }
    }
  __syncthreads();

  // --- theta2 + BN_theta + ReLU -> del ------------------------------------
  for (int otile = 0; otile < 4; ++otile)
    for (int ct = 0; ct < 2; ++ct) {
      v8f acc = wmma_tile64(Wt2, tmp, otile, ct, lane);
#pragma unroll
      for (int r = 0; r < 8; ++r) {
        const int o = otile * 16 + r + 8 * kh;
        float v = acc[r] + cp[1 * C_ + o];
        v = fmaxf(v * cp[6 * C_ + o] + cp[7 * C_ + o], 0.f);
        del[(ct * 16 + m16) * LSTR + o] = v;
      }
    }
  __syncthreads();

  // --- psi: rel = lin_i - (psi(x_j)+b_psi) + delta ------------------------
  for (int otile = 0; otile < 4; ++otile)
    for (int ct = 0; ct < 2; ++ct) {
      v8f acc = wmma_tile64(Wpsi, xj, otile, ct, lane);
#pragma unroll
      for (int r = 0; r < 8; ++r) {
        const int o = otile * 16 + r + 8 * kh;
        const int col = ct * 16 + m16;
        rel[col * LSTR + o] = lini[o] - (acc[r] + cp[2 * C_ + o]) + del[col * LSTR + o];
      }
    }
  __syncthreads();

  // --- gamma1 -> tmp ------------------------------------------------------
  for (int otile = 0; otile < 4; ++otile)
    for (int ct = 0; ct < 2; ++ct) {
      v8f acc = wmma_tile64(Wg1, rel, otile, ct, lane);
#pragma unroll
      for (int r = 0; r < 8; ++r) {
        const int o = otile * 16 + r + 8 * kh;
        tmp[(ct * 16 + m16) * LSTR + o] = acc[r] + cp[4 * C_ + o];
      }
    }
  __syncthreads();

  // --- gamma2 + BN_gamma + ReLU -> rel ------------------------------------
  for (int otile = 0; otile < 4; ++otile)
    for (int ct = 0; ct < 2; ++ct) {
      v8f acc = wmma_tile64(Wg2, tmp, otile, ct, lane);
#pragma unroll
      for (int r = 0; r < 8; ++r) {
        const int o = otile * 16 + r + 8 * kh;
        float v = acc[r] + cp[5 * C_ + o];
        v = fmaxf(v * cp[8 * C_ + o] + cp[9 * C_ + o], 0.f);
        rel[(ct * 16 + m16) * LSTR + o] = v;
      }
    }
  __syncthreads();

  // --- softmax over K (per channel; 2 channels per lane) -> rel = weights -
  for (int oo = lane; oo < C_; oo += 32) {
    float mx = 0.f;  // rel >= 0 after ReLU
    for (int col = 0; col < K_; ++col) mx = fmaxf(mx, rel[col * LSTR + oo]);
    float s = 0.f;
    for (int col = 0; col < K_; ++col) s += __expf(rel[col * LSTR + oo] - mx);
    const float inv = 1.f / s;
    for (int col = 0; col < K_; ++col)
      rel[col * LSTR + oo] = __expf(rel[col * LSTR + oo] - mx) * inv;
  }
  __syncthreads();

  // --- alpha: feats = (alpha(x_j)+b_al+delta)*fmask*weights -> tmp --------
  for (int otile = 0; otile < 4; ++otile)
    for (int ct = 0; ct < 2; ++ct) {
      v8f acc = wmma_tile64(Wal, xj, otile, ct, lane);
#pragma unroll
      for (int r = 0; r < 8; ++r) {
        const int o = otile * 16 + r + 8 * kh;
        const int col = ct * 16 + m16;
        const float feat = acc[r] + cp[3 * C_ + o] + del[col * LSTR + o];
        tmp[col * LSTR + o] = feat * fm[col] * rel[col * LSTR + o];
      }
    }
  __syncthreads();

  // --- reduce over K, write out[b, o, n] ----------------------------------
  for (int oo = lane; oo < C_; oo += 32) {
    float s = 0.f;
    for (int col = 0; col < K_; ++col) s += tmp[col * LSTR + oo];
    out[((size_t)b * C_ + oo) * N_ + n] = s;
  }
}

// ---------------------------------------------------------------------------
// Host launcher
// ---------------------------------------------------------------------------
extern "C" void kernel_launch(void* const* d_in, const int* in_sizes, int n_in,
                              void* d_out, int out_size, void* d_ws, size_t ws_size,
                              hipStream_t stream) {
  const float* qxyz  = (const float*)d_in[0];
  const float* sxyz  = (const float*)d_in[1];
  const int*   qmask = (const int*)d_in[2];
  const int*   smask = (const int*)d_in[3];
  const float* sfeat = (const float*)d_in[4];
  const float* Wt1  = (const float*)d_in[5],  *bt1  = (const float*)d_in[6];
  const float* Wt2  = (const float*)d_in[7],  *bt2  = (const float*)d_in[8];
  const float* Wphi = (const float*)d_in[9],  *bphi = (const float*)d_in[10];
  const float* Wpsi = (const float*)d_in[11], *bpsi = (const float*)d_in[12];
  const float* Wal  = (const float*)d_in[13], *bal  = (const float*)d_in[14];
  const float* Wg1  = (const float*)d_in[15], *bg1  = (const float*)d_in[16];
  const float* Wg2  = (const float*)d_in[17], *bg2  = (const float*)d_in[18];
  const float* g_t  = (const float*)d_in[19], *be_t = (const float*)d_in[20];
  const float* rm_t = (const float*)d_in[21], *rv_t = (const float*)d_in[22];
  const float* g_g  = (const float*)d_in[23], *be_g = (const float*)d_in[24];
  const float* rm_g = (const float*)d_in[25], *rv_g = (const float*)d_in[26];
  float* out = (float*)d_out;

  char* ws = (char*)d_ws;
  float* featT = (float*)ws;                                     // B*M*C f32 (4 MB)
  int*   idxb  = (int*)(ws + (size_t)B_ * M_ * C_ * 4);          // B*N*K i32 (2 MB)
  float* d2b   = (float*)(ws + (size_t)B_ * M_ * C_ * 4
                              + (size_t)B_ * N_ * K_ * 4);       // B*N*K f32 (2 MB)

  (void)hipFuncSetAttribute(reinterpret_cast<const void*>(ball_query_kernel),
                            hipFuncAttributeMaxDynamicSharedMemorySize, 131072);
  (void)hipFuncSetAttribute(reinterpret_cast<const void*>(pt_fused_kernel),
                            hipFuncAttributeMaxDynamicSharedMemorySize, 146432);

  transpose_feat<<<dim3(M_ / 32, C_ / 32, B_), dim3(32, 8), 0, stream>>>(sfeat, featT);

  ball_query_kernel<<<dim3((B_ * N_) / 4), 128, 131072, stream>>>(qxyz, sxyz, smask,
                                                                  idxb, d2b);

  pt_fused_kernel<<<dim3((B_ * N_) / 4), 128, 146432, stream>>>(
      qxyz, sxyz, qmask, featT, idxb, d2b,
      Wt1, bt1, Wt2, bt2, Wphi, bphi, Wpsi, bpsi, Wal, bal, Wg1, bg1, Wg2, bg2,
      g_t, be_t, rm_t, rv_t, g_g, be_g, rm_g, rv_g, out);
}